// RobustTripletLoss_LCNL_76321568850073
// MI455X (gfx1250) — compile-verified
//
#include <hip/hip_runtime.h>
#include <math.h>

#define NN 4096
#define DD 1024
#define KK 32
#define MARGINF 0.3f

typedef __attribute__((ext_vector_type(2))) float v2f;
typedef __attribute__((ext_vector_type(8))) float v8f;

__device__ __forceinline__ bool pair_lt(float d1, int i1, float d2, int i2) {
  return (d1 < d2) || ((d1 == d2) && (i1 < i2));
}

// Sorted-insert via predicated swap chain; keeps L smallest (asc) / largest
// (desc) with index tie-break matching jnp.argsort's stable order. Exact
// regardless of insertion order (lexicographic comparator).
template<int L>
__device__ __forceinline__ void insert_asc(float (&vd)[L], int (&vi)[L], float d, int i) {
#pragma unroll
  for (int j = 0; j < L; ++j) {
    if (pair_lt(d, i, vd[j], vi[j])) {
      float td = vd[j]; int ti = vi[j];
      vd[j] = d; vi[j] = i; d = td; i = ti;
    }
  }
}

template<int L>
__device__ __forceinline__ void insert_desc(float (&vd)[L], int (&vi)[L], float d, int i) {
#pragma unroll
  for (int j = 0; j < L; ++j) {
    bool b = (d > vd[j]) || ((d == vd[j]) && (i < vi[j]));
    if (b) {
      float td = vd[j]; int ti = vi[j];
      vd[j] = d; vi[j] = i; d = td; i = ti;
    }
  }
}

// Kernel 1: squared row norms (one wave32 per row).
__global__ void sqnorm_kernel(const float* __restrict__ in, float* __restrict__ sq) {
  int row = blockIdx.x;
  const float* p = in + (size_t)row * DD;
  float s = 0.f;
  for (int i = threadIdx.x; i < DD; i += 32) { float v = p[i]; s += v * v; }
  for (int off = 16; off > 0; off >>= 1) s += __shfl_down(s, off, 32);
  if (threadIdx.x == 0) sq[row] = s;
}

// Kernel 2: fused WMMA distance tiles + per-row top-k selection + epilogue.
// Block = 128 threads (4 wave32), owns 16 rows. Each wave covers a contiguous
// quarter of the 256 column tiles, processing a 16x64 strip (4 accumulators)
// per pass so one A fragment feeds 4 WMMAs (5 loads / 4 WMMAs).
__global__ void __launch_bounds__(128)
triplet_kernel(const float* __restrict__ in, const float* __restrict__ pred,
               const int* __restrict__ targets, const float* __restrict__ prob,
               const float* __restrict__ thr_p, const float* __restrict__ sq,
               float* __restrict__ accum) {
  __shared__ float sqr[16];
  __shared__ int   tgl[4][16];
  __shared__ float dt[4][16 * 17];
  __shared__ float m_nd[4][16][12];
  __shared__ int   m_ni[4][16][12];
  __shared__ float m_pd[4][16][7];
  __shared__ int   m_pi[4][16][7];

  const int r0   = blockIdx.x * 16;
  const int wave = threadIdx.x >> 5;
  const int lane = threadIdx.x & 31;
  const int half = lane >> 4;
  const int lr   = lane & 15;

  if (threadIdx.x < 16) sqr[threadIdx.x] = sq[r0 + threadIdx.x];
  __syncthreads();

  // Candidate lists (meaningful in lanes 0..15, one row per lane).
  float nd[12]; int ni[12];
  float pd[7];  int pi[7];
#pragma unroll
  for (int j = 0; j < 12; ++j) { nd[j] = 3.4e38f; ni[j] = 0x7fffffff; }
#pragma unroll
  for (int j = 0; j < 7; ++j)  { pd[j] = -3.4e38f; pi[j] = 0x7fffffff; }

  const int myrow = r0 + lr;
  const int mycls = targets[myrow];

  // A fragment (16x4 f32): lanes 0-15 hold K0,K1; lanes 16-31 hold K2,K3.
  const float* arow = in + (size_t)myrow * DD + 2 * half;

  const int tilesPerWave = (NN / 16) / 4;  // 64 column tiles per wave
  for (int g = 0; g < tilesPerWave; g += 4) {
    const int ct0 = wave * tilesPerWave + g;
    const int c0  = ct0 * 16;
    // B = A^T of the same matrix -> identical per-lane addressing pattern.
    const float* b0 = in + (size_t)(c0 + lr) * DD + 2 * half;

    v8f acc[4];
#pragma unroll
    for (int t = 0; t < 4; ++t) acc[t] = (v8f){0.f,0.f,0.f,0.f,0.f,0.f,0.f,0.f};

#pragma unroll 2
    for (int k0 = 0; k0 < DD; k0 += 4) {
      v2f a = *(const v2f*)(arow + k0);
#pragma unroll
      for (int t = 0; t < 4; ++t) {
        v2f b = *(const v2f*)(b0 + (size_t)(16 * t) * DD + k0);
        acc[t] = __builtin_amdgcn_wmma_f32_16x16x4_f32(
            false, a, false, b, (short)0, acc[t], false, false);
      }
    }

#pragma unroll
    for (int t = 0; t < 4; ++t) {
      const int cc0 = c0 + 16 * t;
      const float sqc = sq[cc0 + lr];
      if (lane < 16) tgl[wave][lane] = targets[cc0 + lane];
      // dist[m][n] = sqrt(max(sq_m + sq_n - 2*gram, 1e-12)); C layout: M=i+8*half, N=lr.
#pragma unroll
      for (int i = 0; i < 8; ++i) {
        int m = i + 8 * half;
        float dd = sqr[m] + sqc - 2.0f * acc[t][i];
        dd = sqrtf(fmaxf(dd, 1e-12f));
        dt[wave][m * 17 + lr] = dd;
      }
      if (lane < 16) {
        for (int c = 0; c < 16; ++c) {
          float dd = dt[wave][lane * 17 + c];
          int idx = cc0 + c;
          if (tgl[wave][c] == mycls) insert_desc<7>(pd, pi, dd, idx);
          else                       insert_asc<12>(nd, ni, dd, idx);
        }
      }
    }
  }

  if (lane < 16) {
#pragma unroll
    for (int j = 0; j < 12; ++j) { m_nd[wave][lane][j] = nd[j]; m_ni[wave][lane][j] = ni[j]; }
#pragma unroll
    for (int j = 0; j < 7; ++j)  { m_pd[wave][lane][j] = pd[j]; m_pi[wave][lane][j] = pi[j]; }
  }
  __syncthreads();

  if (wave == 0 && lane < 16) {
    // Exact merge of 4 sorted partial lists per row.
    float fnd[12]; int fni[12]; float fpd[7]; int fpi[7];
#pragma unroll
    for (int j = 0; j < 12; ++j) { fnd[j] = 3.4e38f; fni[j] = 0x7fffffff; }
#pragma unroll
    for (int j = 0; j < 7; ++j)  { fpd[j] = -3.4e38f; fpi[j] = 0x7fffffff; }
    for (int w = 0; w < 4; ++w) {
      for (int j = 0; j < 12; ++j) insert_asc<12>(fnd, fni, m_nd[w][lane][j], m_ni[w][lane][j]);
      for (int j = 0; j < 7; ++j)  insert_desc<7>(fpd, fpi, m_pd[w][lane][j], m_pi[w][lane][j]);
    }

    const float thr = thr_p[0];
    const int row = r0 + lane;
    const bool conf = prob[row] >= thr;
    const int hp = fpi[0], hn = fni[0];
    const float d_ap = fpd[0], d_an = fnd[0];
    const bool cp = prob[hp] >= thr;
    const bool cn = prob[hn] >= thr;

    // fn = (argmax(prediction[hn]) == targets[row]); first-max tie-break.
    const float* pr = pred + (size_t)hn * KK;
    float best = pr[0]; int bi = 0;
    for (int j = 1; j < KK; ++j) { float v = pr[j]; if (v > best) { best = v; bi = j; } }
    const bool fn = (bi == mycls);

    int sel_n = 10;
    for (int j = 9; j >= 0; --j) if (prob[fni[1 + j]] >= thr) sel_n = j;
    const float d_nn = fnd[1 + sel_n];
    int sel_p = 5;
    for (int j = 4; j >= 0; --j) if (prob[fpi[1 + j]] >= thr) sel_p = j;
    const float d_np = fpd[1 + sel_p];

    const float e1 = expf(d_ap), e2 = expf(d_an);
    const float w1 = (e1 * d_ap + e2 * d_an) / (e1 + e2);
    const float f1 = expf(-d_ap), f2 = expf(-d_an);
    const float w0 = (f1 * d_ap + f2 * d_an) / (f1 + f2 + 1e-6f);

    const bool case_b  = cp && !cn && fn;
    const bool case_cd = !cp && (cn || !fn);
    const bool inv     = !cp && !cn && fn;
    const float ap = case_b ? w1 : (case_cd ? d_np : d_ap);
    const float an = case_b ? d_nn : (case_cd ? w0 : d_an);
    const float per = inv ? fmaxf(an - ap + MARGINF, 0.f)
                          : fmaxf(ap - an + MARGINF, 0.f);
    if (conf) {
      atomicAdd(&accum[0], per);
      atomicAdd((int*)accum + 2, 1);
      if (an >= ap) atomicAdd((int*)accum + 1, 1);
    }
  }
}

__global__ void finalize_kernel(const float* __restrict__ accum, float* __restrict__ out) {
  float ls = accum[0];
  int corr = ((const int*)accum)[1];
  int cnt  = ((const int*)accum)[2];
  out[0] = (cnt > 0) ? ls / (float)cnt : 0.0f;
  out[1] = (float)corr;
  out[2] = (float)cnt;
}

extern "C" void kernel_launch(void* const* d_in, const int* in_sizes, int n_in,
                              void* d_out, int out_size, void* d_ws, size_t ws_size,
                              hipStream_t stream) {
  (void)in_sizes; (void)n_in; (void)out_size; (void)ws_size;
  const float* inputs    = (const float*)d_in[0];
  const float* pred      = (const float*)d_in[1];
  const int*   targets   = (const int*)d_in[2];
  // d_in[3] = true_targets (unused by the reference)
  const float* prob      = (const float*)d_in[4];
  const float* threshold = (const float*)d_in[5];

  float* sq    = (float*)d_ws;      // NN floats
  float* accum = sq + NN;           // [0]=loss_sum(f32), [1]=correct(i32), [2]=cnt(i32)

  hipMemsetAsync(accum, 0, 3 * sizeof(float), stream);
  sqnorm_kernel<<<NN, 32, 0, stream>>>(inputs, sq);
  triplet_kernel<<<NN / 16, 128, 0, stream>>>(inputs, pred, targets, prob,
                                              threshold, sq, accum);
  finalize_kernel<<<1, 1, 0, stream>>>(accum, (float*)d_out);
}